// HistogramLoss_13623636263533
// MI455X (gfx1250) — compile-verified
//
#include <hip/hip_runtime.h>
#include <stdint.h>

#define PIX 16384      // H*W per channel
#define NB  255        // histogram bins

typedef __attribute__((ext_vector_type(2))) float v2f;
typedef __attribute__((ext_vector_type(8))) float v8f;
typedef __attribute__((ext_vector_type(4))) unsigned int v4u;
typedef __attribute__((ext_vector_type(4))) int v4i;
typedef __attribute__((ext_vector_type(8))) int v8i;

// ---------------------------------------------------------------------------
// Block-wide reduce (op: 0=add, 1=min, 2=max). Deterministic fixed-order tree.
// ---------------------------------------------------------------------------
__device__ __forceinline__ float block_reduce(float v, float* red, int tid, int op) {
  __syncthreads();
  red[tid] = v;
  __syncthreads();
  for (int s = 128; s > 0; s >>= 1) {
    if (tid < s) {
      float a = red[tid], b = red[tid + s];
      red[tid] = (op == 0) ? (a + b) : (op == 1 ? fminf(a, b) : fmaxf(a, b));
    }
    __syncthreads();
  }
  float r = red[0];
  __syncthreads();
  return r;
}

// ---------------------------------------------------------------------------
// 256-wide inclusive scan on the matrix core: view h as 16x16 row-major H.
// Row-prefix P = H x U (upper-triangular ones) via 4 chained
// V_WMMA_F32_16X16X4_F32. Row offsets fixed up by caller. Wave32, EXEC all 1s.
// ---------------------------------------------------------------------------
__device__ __forceinline__ void wmma_scan_rows(const float* H, float* OUT, int lane) {
  const int m    = lane & 15;
  const int half = lane >> 4;
  v8f c = {};
#pragma unroll
  for (int kb = 0; kb < 4; ++kb) {
    const int k0 = 4 * kb + 2 * half;
    v2f a, b;
    a.x = H[16 * m + k0];
    a.y = H[16 * m + k0 + 1];
    b.x = (k0     <= m) ? 1.0f : 0.0f;
    b.y = (k0 + 1 <= m) ? 1.0f : 0.0f;
    c = __builtin_amdgcn_wmma_f32_16x16x4_f32(false, a, false, b, (short)0, c, false, false);
  }
#pragma unroll
  for (int v = 0; v < 8; ++v) {
    const int r = half * 8 + v;
    OUT[16 * r + m] = c[v];
  }
}

// ---------------------------------------------------------------------------
// TDM: stage one contiguous 16384-float row (64 KB) from global into LDS.
// D# per ISA cdna5 §8: group0 = {count=1 | lds_addr | global_addr | type=2},
// group1 = {data_size=4B, tensor_dim0=tile_dim0=16384, tensor_dim1=tile_dim1=1,
//           tensor_dim0_stride=16384}. Groups 2/3 unused (<=2D).
// ---------------------------------------------------------------------------
__device__ __forceinline__ void tdm_load_row(const float* gsrc, unsigned lds_off) {
  unsigned long long ga = (unsigned long long)(uintptr_t)gsrc;
  v4u g0;
  g0.x = 1u;                                   // count=1, user mode
  g0.y = lds_off;                              // lds_addr
  g0.z = (unsigned)(ga & 0xffffffffu);         // global_addr[31:0]
  g0.w = (unsigned)((ga >> 32) & 0x01ffffffu)  // global_addr[56:32]
         | (2u << 30);                         // type=2 ("image")
  v8i g1;
  g1[0] = (int)(2u << 16);                     // data_size = 4 bytes
  g1[1] = (int)(16384u << 16);                 // tensor_dim0 = 16384 (low 16 bits)
  g1[2] = (int)(1u << 16);                     // tensor_dim0 hi=0 | tensor_dim1 = 1
  g1[3] = (int)(16384u << 16);                 // tensor_dim1 hi=0 | tile_dim0 = 16384
  g1[4] = 1;                                   // tile_dim1 = 1, tile_dim2 = 0
  g1[5] = 16384;                               // tensor_dim0_stride = 16384
  g1[6] = 0;
  g1[7] = 0;
  v4i gz = {0, 0, 0, 0};
#if defined(__clang_major__) && (__clang_major__ >= 23)
  v8i gz8 = {0, 0, 0, 0, 0, 0, 0, 0};
  __builtin_amdgcn_tensor_load_to_lds(g0, g1, gz, gz, gz8, 0);
#else
  __builtin_amdgcn_tensor_load_to_lds(g0, g1, gz, gz, 0);
#endif
}

// ---------------------------------------------------------------------------
// One workgroup per (batch,channel). 256 threads = 8 waves.
// ---------------------------------------------------------------------------
__global__ __launch_bounds__(256)
void histmatch_kernel(const float* __restrict__ in, const float* __restrict__ tg,
                      float* __restrict__ partial) {
  __shared__ float sU[PIX];        // 64 KB: input channel
  __shared__ float sT[PIX];        // 64 KB: target channel
  __shared__ float shist[256];
  __shared__ float thist[256];
  __shared__ float scdf[256];
  __shared__ float tcdf[256];
  __shared__ float red[256];
  __shared__ int   iLUT[256];      // inverse of quantized target CDF
  __shared__ float offS[16];
  __shared__ float offT[16];

  const int tid  = threadIdx.x;
  const int lane = tid & 31;
  const int wid  = tid >> 5;
  const long long base = (long long)blockIdx.x * PIX;
  const float* gin = in + base;
  const float* gtg = tg + base;

  // ---- Phase 0: stage both channels into LDS, one HBM pass each.
  //      sT: Tensor Data Mover (one DMA descriptor, wave 0 issues).
  //      sU: async global->LDS B128 copies (all waves).
  if (wid == 0) {
    tdm_load_row(gtg, (unsigned)(uintptr_t)&sT[0]);
  }
  for (int i = tid * 4; i < PIX; i += 256 * 4) {
    unsigned uoff = (unsigned)(uintptr_t)&sU[i];
    unsigned long long ga = (unsigned long long)(uintptr_t)(gin + i);
    asm volatile("global_load_async_to_lds_b128 %0, %1, off" :: "v"(uoff), "v"(ga) : "memory");
  }
  asm volatile("s_wait_asynccnt 0" ::: "memory");
  if (wid == 0) __builtin_amdgcn_s_wait_tensorcnt(0);
  __syncthreads();

  // ---- Phase 1: per-channel min/max of quantized values (float4 LDS reads)
  float mnS = 3.0e38f, mxS = -3.0e38f, mnT = 3.0e38f, mxT = -3.0e38f;
  for (int i = tid * 4; i < PIX; i += 1024) {
    float4 us = *(const float4*)&sU[i];
    float4 ts = *(const float4*)&sT[i];
    float uu[4] = {us.x, us.y, us.z, us.w};
    float tt[4] = {ts.x, ts.y, ts.z, ts.w};
#pragma unroll
    for (int q = 0; q < 4; ++q) {
      float sv = floorf((uu[q] + 1.0f) * 127.5f);
      float tv = floorf((tt[q] + 1.0f) * 127.5f);
      mnS = fminf(mnS, sv); mxS = fmaxf(mxS, sv);
      mnT = fminf(mnT, tv); mxT = fmaxf(mxT, tv);
    }
  }
  const float loS = block_reduce(mnS, red, tid, 1);
  const float hiS = block_reduce(mxS, red, tid, 2);
  const float loT = block_reduce(mnT, red, tid, 1);
  const float hiT = block_reduce(mxT, red, tid, 2);
  const float wS = hiS - loS;
  const float wT = hiT - loT;
  const float stepT  = wT * (1.0f / 255.0f);
  const float edgeHi = loT + 254.0f * stepT;   // xp[n-1] of edges_t[:-1]

  // ---- Phase 2: histograms via LDS float atomics (ds_add_f32)
  shist[tid] = 0.0f;
  thist[tid] = 0.0f;
  __syncthreads();
  const float invWS = (wS > 0.0f) ? 255.0f / wS : 0.0f;
  const float invWT = (wT > 0.0f) ? 255.0f / wT : 0.0f;
  for (int i = tid * 4; i < PIX; i += 1024) {
    float4 us = *(const float4*)&sU[i];
    float4 ts = *(const float4*)&sT[i];
    float uu[4] = {us.x, us.y, us.z, us.w};
    float tt[4] = {ts.x, ts.y, ts.z, ts.w};
#pragma unroll
    for (int q = 0; q < 4; ++q) {
      float sv = floorf((uu[q] + 1.0f) * 127.5f);
      float tv = floorf((tt[q] + 1.0f) * 127.5f);
      int bs = min(max((int)floorf((sv - loS) * invWS), 0), NB - 1);
      int bt = min(max((int)floorf((tv - loT) * invWT), 0), NB - 1);
      atomicAdd(&shist[bs], 1.0f);
      atomicAdd(&thist[bt], 1.0f);
    }
  }
  __syncthreads();

  // ---- Phase 3: CDF = 256-wide scan on the matrix pipe (2 waves, 8 WMMAs)
  if (wid == 0)      wmma_scan_rows(shist, scdf, lane);
  else if (wid == 1) wmma_scan_rows(thist, tcdf, lane);
  __syncthreads();
  if (tid == 0) {
    float run = 0.0f;
    for (int r = 0; r < 16; ++r) { offS[r] = run; run += scdf[16 * r + 15]; }
  } else if (tid == 1) {
    float run = 0.0f;
    for (int r = 0; r < 16; ++r) { offT[r] = run; run += tcdf[16 * r + 15]; }
  }
  __syncthreads();
  {
    float cs = scdf[tid] + offS[tid >> 4];
    float ct = tcdf[tid] + offT[tid >> 4];
    __syncthreads();
    scdf[tid] = cs;
    tcdf[tid] = ct;
  }
  __syncthreads();
  const float totS = scdf[NB - 1];
  const float totT = tcdf[NB - 1];
  {
    float qs = floorf(255.0f * scdf[tid] / totS);   // floor == uint8 cast
    float qt = floorf(255.0f * tcdf[tid] / totT);
    __syncthreads();
    scdf[tid] = qs;
    tcdf[tid] = qt;
  }
  __syncthreads();

  // ---- Phase 3b: inverse LUT of quantized target CDF.
  // iLUT[v] = largest k in [0,254] with tcdf[k] <= v  (tcdf entries are exact
  // integers in [0,255], so searchsorted collapses to a table lookup).
  iLUT[tid] = 0;
  __syncthreads();
  if (tid < NB) atomicMax(&iLUT[min((int)tcdf[tid], 255)], tid);
  __syncthreads();
  for (int s = 1; s < 256; s <<= 1) {      // inclusive prefix-max, 8 steps
    int a = iLUT[tid];
    int b = (tid >= s) ? iLUT[tid - s] : 0;
    __syncthreads();
    iLUT[tid] = max(a, b);
    __syncthreads();
  }

  // ---- Phase 4: double interp + MSE accumulation (LDS-resident)
  const float cdfS0 = scdf[0],  cdfSn = scdf[NB - 1];
  const float cdfT0 = tcdf[0],  cdfTn = tcdf[NB - 1];
  const float invStepT = (stepT > 0.0f) ? 1.0f / stepT : 0.0f;
  float acc = 0.0f;
  for (int i = tid * 4; i < PIX; i += 1024) {
    float4 us = *(const float4*)&sU[i];
    float uu[4] = {us.x, us.y, us.z, us.w};
#pragma unroll
    for (int q = 0; q < 4; ++q) {
      const float u = uu[q];
      const float x = floorf((u + 1.0f) * 127.5f);
      // interp #1: xp = uniform target edges, fp = source cdf (closed form idx)
      float im2;
      if (x <= loT)          im2 = cdfS0;
      else if (x >= edgeHi)  im2 = cdfSn;
      else {
        int j = min(max((int)floorf((x - loT) * invStepT), 0), NB - 2);
        float x0  = loT + (float)j * stepT;
        float t01 = (x - x0) * invStepT;
        float f0  = scdf[j];
        im2 = f0 + t01 * (scdf[j + 1] - f0);
      }
      // interp #2: xp = target cdf (monotone ints) -> inverse LUT lookup
      float y;
      if (im2 <= cdfT0)      y = loT;
      else if (im2 >= cdfTn) y = edgeHi;
      else {
        int j = min(iLUT[(int)im2], NB - 2);
        float c0 = tcdf[j];
        float d  = tcdf[j + 1] - c0;
        float t01 = (d > 0.0f) ? (im2 - c0) / d : 0.0f;
        y = loT + ((float)j + t01) * stepT;
      }
      const float hm = floorf(y) * (2.0f / 255.0f) - 1.0f;
      const float dd = u - hm;
      acc = fmaf(dd, dd, acc);
    }
  }
  const float bsum = block_reduce(acc, red, tid, 0);
  if (tid == 0) partial[blockIdx.x] = bsum;
}

// ---------------------------------------------------------------------------
// Deterministic final reduction of per-channel partial sums -> mean.
// ---------------------------------------------------------------------------
__global__ __launch_bounds__(256)
void final_reduce_kernel(const float* __restrict__ partial, float* __restrict__ out,
                         int n, float invN) {
  __shared__ float red[256];
  float a = 0.0f;
  for (int i = threadIdx.x; i < n; i += 256) a += partial[i];
  red[threadIdx.x] = a;
  __syncthreads();
  for (int s = 128; s > 0; s >>= 1) {
    if (threadIdx.x < s) red[threadIdx.x] += red[threadIdx.x + s];
    __syncthreads();
  }
  if (threadIdx.x == 0) out[0] = red[0] * invN;
}

extern "C" void kernel_launch(void* const* d_in, const int* in_sizes, int n_in,
                              void* d_out, int out_size, void* d_ws, size_t ws_size,
                              hipStream_t stream) {
  const float* in = (const float*)d_in[0];
  const float* tg = (const float*)d_in[1];
  float* out     = (float*)d_out;
  float* partial = (float*)d_ws;          // nch floats of scratch
  const long long n = (long long)in_sizes[0];
  const int nch = (int)(n / PIX);         // B*C = 2048 for the reference shape
  if (nch <= 0) return;
  histmatch_kernel<<<nch, 256, 0, stream>>>(in, tg, partial);
  const float invN = 1.0f / (float)n;
  final_reduce_kernel<<<1, 256, 0, stream>>>(partial, out, nch, invN);
}